// MemoryBank_85770496901144
// MI455X (gfx1250) — compile-verified
//
#include <hip/hip_runtime.h>
#include <stdint.h>

#define C_CLASSES 1000
#define N_SLOTS   256
#define D_FEAT    512
#define B_BATCH   4096

typedef unsigned int v4u  __attribute__((ext_vector_type(4)));
typedef int          v8i_t __attribute__((ext_vector_type(8)));
typedef int          v4i_t __attribute__((ext_vector_type(4)));

// ---------------------------------------------------------------------------
// TDM wrappers (ROCm 7.2 clang-22: 5 args; amdgpu-toolchain clang-23: 6 args)
// ---------------------------------------------------------------------------
#if defined(__has_builtin) && __has_builtin(__builtin_amdgcn_tensor_load_to_lds)
#define HAVE_TDM 1
__device__ __forceinline__ void tdm_load(v4u g0, v8i_t g1, v4i_t g2, v4i_t g3) {
#if __clang_major__ >= 23
  v8i_t z = (v8i_t)0;
  __builtin_amdgcn_tensor_load_to_lds(g0, g1, g2, g3, z, 0);
#else
  __builtin_amdgcn_tensor_load_to_lds(g0, g1, g2, g3, 0);
#endif
}
__device__ __forceinline__ void tdm_store(v4u g0, v8i_t g1, v4i_t g2, v4i_t g3) {
#if __clang_major__ >= 23
  v8i_t z = (v8i_t)0;
  __builtin_amdgcn_tensor_store_from_lds(g0, g1, g2, g3, z, 0);
#else
  __builtin_amdgcn_tensor_store_from_lds(g0, g1, g2, g3, 0);
#endif
}
__device__ __forceinline__ void tdm_wait0() {
  __builtin_amdgcn_s_wait_tensorcnt(0);
}
#else
#define HAVE_TDM 0
#warning "tensor_load_to_lds builtin not available; TDM path disabled"
#endif

// ---------------------------------------------------------------------------
// K3 (defined FIRST so the disasm snippet shows it): bandwidth kernel.
// One wave per class. TDM gather-mode load pulls 16 permuted rows (2 KB each)
// per descriptor into LDS; tensor_store streams the contiguous sorted tile to
// the output. Double-buffered 2x32KB: store k overlaps load k+1 (same-wave
// tensor ops complete in order, so s_wait_tensorcnt 0 after load_k is the
// tightest correct wait). The single "new feature" row is patched afterwards.
// ---------------------------------------------------------------------------
__global__ void __launch_bounds__(32)
gather_tdm_kernel(const float* __restrict__ memory,
                  const float* __restrict__ batch_features,
                  const int*   __restrict__ last_pos,
                  const int*   __restrict__ order,
                  float* __restrict__ out) {
  __shared__ float buf[2][16 * D_FEAT];   // 2 x 32 KB
  const int c  = blockIdx.x;
  const int lp = last_pos[c];
  const bool upd = (lp >= 0);
  const int* ord = order + (size_t)c * N_SLOTS;
  const uint64_t srcBase = (uint64_t)(uintptr_t)(memory + (size_t)c * N_SLOTS * D_FEAT);
  const uint64_t dstBase = (uint64_t)(uintptr_t)(out    + (size_t)c * N_SLOTS * D_FEAT);
  int featPos = -1;

  for (int k = 0; k < 16; ++k) {
    // pack 16 x 16-bit source-row indices (fold the +1 slot shift in here)
    unsigned idxpack[8] = {0, 0, 0, 0, 0, 0, 0, 0};
    for (int t = 0; t < 16; ++t) {
      int j = ord[k * 16 + t];
      unsigned srcRow;
      if (upd) {
        if (j == N_SLOTS - 1) { featPos = k * 16 + t; srcRow = 0u; }  // dummy, patched later
        else                  { srcRow = (unsigned)(j + 1); }
      } else {
        srcRow = (unsigned)j;
      }
      idxpack[t >> 1] |= (srcRow & 0xFFFFu) << (16 * (t & 1));
    }
    const unsigned ldsAddr = (unsigned)(uintptr_t)(&buf[k & 1][0]);  // addr[31:0] = LDS offset

#if HAVE_TDM
    // ---- gather-mode load descriptor ----
    v4u g0l;
    g0l.x = 0x80000001u;                                  // count=1, gather_mode=1, 16-bit idx
    g0l.y = ldsAddr;                                      // lds_addr
    g0l.z = (unsigned)(srcBase & 0xFFFFFFFFull);          // global_addr[31:0]
    g0l.w = (unsigned)((srcBase >> 32) & 0x01FFFFFFull) | (2u << 30);  // addr[56:32], type=2
    v8i_t g1;
    g1[0] = (int)(2u << 16);                              // data_size = 4 B
    g1[1] = (int)(((unsigned)D_FEAT  & 0xFFFFu) << 16);   // tensor_dim0 = 512
    g1[2] = (int)(((unsigned)N_SLOTS & 0xFFFFu) << 16);   // tensor_dim1 = 256
    g1[3] = (int)(((unsigned)D_FEAT  & 0xFFFFu) << 16);   // tile_dim0 = 512
    g1[4] = 16;                                           // tile_dim1 = #valid indices
    g1[5] = D_FEAT;                                       // tensor_dim0_stride = 512
    g1[6] = 0;
    g1[7] = 0;
    v4i_t g2, g3;
    g2[0] = (int)idxpack[0]; g2[1] = (int)idxpack[1];
    g2[2] = (int)idxpack[2]; g2[3] = (int)idxpack[3];
    g3[0] = (int)idxpack[4]; g3[1] = (int)idxpack[5];
    g3[2] = (int)idxpack[6]; g3[3] = (int)idxpack[7];
    tdm_load(g0l, g1, g2, g3);
    tdm_wait0();   // in-order completion: also retires the previous chunk's store

    // ---- contiguous 16x512 store descriptor ----
    const uint64_t dstAddr = dstBase + (uint64_t)k * 16u * D_FEAT * 4u;
    v4u g0s;
    g0s.x = 1u;                                           // count=1, gather off
    g0s.y = ldsAddr;
    g0s.z = (unsigned)(dstAddr & 0xFFFFFFFFull);
    g0s.w = (unsigned)((dstAddr >> 32) & 0x01FFFFFFull) | (2u << 30);
    v8i_t g1s;
    g1s[0] = (int)(2u << 16);                             // data_size = 4 B
    g1s[1] = (int)(((unsigned)D_FEAT & 0xFFFFu) << 16);   // tensor_dim0 = 512
    g1s[2] = (int)((16u & 0xFFFFu) << 16);                // tensor_dim1 = 16 (tile == tensor)
    g1s[3] = (int)(((unsigned)D_FEAT & 0xFFFFu) << 16);   // tile_dim0 = 512
    g1s[4] = 16;                                          // tile_dim1 = 16
    g1s[5] = D_FEAT;                                      // stride = 512
    g1s[6] = 0;
    g1s[7] = 0;
    v4i_t zz = (v4i_t)0;
    tdm_store(g0s, g1s, zz, zz);   // overlaps with next iteration's load
#else
    // fallback: plain wide-vector gather, one wave copies 16 rows
    for (int t = 0; t < 16; ++t) {
      int j = ord[k * 16 + t];
      unsigned srcRow = upd ? ((j == N_SLOTS - 1) ? 0u : (unsigned)(j + 1)) : (unsigned)j;
      const float4* s = (const float4*)(memory + ((size_t)c * N_SLOTS + srcRow) * D_FEAT);
      float4* dptr = (float4*)(out + ((size_t)c * N_SLOTS + k * 16 + t) * D_FEAT);
      for (int q = threadIdx.x; q < D_FEAT / 4; q += 32) dptr[q] = s[q];
    }
#endif
  }
#if HAVE_TDM
  tdm_wait0();
#endif
  // patch the appended-feature row (mem2[:, N-1] for updated classes)
  if (upd && featPos >= 0) {
    const float4* s = (const float4*)(batch_features + (size_t)lp * D_FEAT);
    float4* dptr = (float4*)(out + ((size_t)c * N_SLOTS + featPos) * D_FEAT);
    for (int q = threadIdx.x; q < D_FEAT / 4; q += 32) dptr[q] = s[q];
  }
}

// ---------------------------------------------------------------------------
// K0: init last_pos to -1
// ---------------------------------------------------------------------------
__global__ void init_lastpos_kernel(int* __restrict__ last_pos) {
  int i = blockIdx.x * blockDim.x + threadIdx.x;
  if (i < C_CLASSES) last_pos[i] = -1;
}

// ---------------------------------------------------------------------------
// K1: per-batch-row argmax (first occurrence of max) + validity + segment-max
// of batch positions via atomicMax  (jax segment_max over pos == last write)
// ---------------------------------------------------------------------------
__global__ void argmax_push_kernel(const int*   __restrict__ batch_targets,
                                   const float* __restrict__ batch_confidences,
                                   const int*   __restrict__ selected_mask,
                                   const float* __restrict__ confidences,
                                   int* __restrict__ last_pos) {
  __shared__ int sval[256];
  __shared__ int sidx[256];
  const int b = blockIdx.x;
  const int t = threadIdx.x;
  const int* row = batch_targets + (size_t)b * C_CLASSES;
  int bestV = -2147483647 - 1;
  int bestI = 0x7FFFFFFF;
  for (int col = t; col < C_CLASSES; col += 256) {
    int v = row[col];
    if (v > bestV) { bestV = v; bestI = col; }   // strict > keeps first occurrence
  }
  sval[t] = bestV;
  sidx[t] = bestI;
  __syncthreads();
  for (int s = 128; s > 0; s >>= 1) {
    if (t < s) {
      int v2 = sval[t + s], i2 = sidx[t + s];
      if (v2 > sval[t] || (v2 == sval[t] && i2 < sidx[t])) { sval[t] = v2; sidx[t] = i2; }
    }
    __syncthreads();
  }
  if (t == 0) {
    int cls = sidx[0];
    bool valid = (selected_mask[b] != 0) &&
                 (batch_confidences[b] > confidences[(size_t)cls * N_SLOTS + (N_SLOTS - 1)]);
    if (valid) atomicMax(&last_pos[cls], b);
  }
}

// ---------------------------------------------------------------------------
// K2: per-class stable descending argsort of 256 confidences.
// Composite 64-bit key = (~orderable(conf) << 32) | slot  -> unique keys,
// ascending bitonic sort reproduces stable argsort(-conf) exactly.
// ---------------------------------------------------------------------------
__global__ void sort_kernel(const float* __restrict__ confidences,
                            const float* __restrict__ batch_confidences,
                            const int*   __restrict__ last_pos,
                            int* __restrict__ order) {
  __shared__ unsigned long long keys[N_SLOTS];
  const int c = blockIdx.x;
  const int i = threadIdx.x;
  const int lp = last_pos[c];
  float cf = confidences[(size_t)c * N_SLOTS + i];
  if (lp >= 0 && i == N_SLOTS - 1) cf = batch_confidences[lp];   // conf2[:, -1]
  unsigned u = __float_as_uint(cf);
  u = (u & 0x80000000u) ? ~u : (u | 0x80000000u);  // order-preserving (ascending)
  unsigned d = ~u;                                  // descending on conf
  keys[i] = ((unsigned long long)d << 32) | (unsigned)i;
  __syncthreads();
  for (int k = 2; k <= N_SLOTS; k <<= 1) {
    for (int j = k >> 1; j > 0; j >>= 1) {
      int partner = i ^ j;
      if (partner > i) {
        bool ascend = ((i & k) == 0);
        unsigned long long a = keys[i], b2 = keys[partner];
        if ((a > b2) == ascend) { keys[i] = b2; keys[partner] = a; }
      }
      __syncthreads();
    }
  }
  order[(size_t)c * N_SLOTS + i] = (int)(keys[i] & 0xFFFFFFFFu);
}

// ---------------------------------------------------------------------------
extern "C" void kernel_launch(void* const* d_in, const int* in_sizes, int n_in,
                              void* d_out, int out_size, void* d_ws, size_t ws_size,
                              hipStream_t stream) {
  const float* batch_features    = (const float*)d_in[0];   // [B, D]
  const int*   batch_targets     = (const int*)  d_in[1];   // [B, C]
  const float* batch_confidences = (const float*)d_in[2];   // [B]
  const int*   selected_mask     = (const int*)  d_in[3];   // [B]
  const float* memory            = (const float*)d_in[4];   // [C, N, D]
  const float* confidences       = (const float*)d_in[5];   // [C, N]
  float* out = (float*)d_out;                               // [C, N, D]

  char* ws = (char*)d_ws;
  int* last_pos = (int*)(ws);                               // C ints
  int* order    = (int*)(ws + 8192);                        // C*N ints (~1 MB)

  init_lastpos_kernel<<<(C_CLASSES + 255) / 256, 256, 0, stream>>>(last_pos);
  argmax_push_kernel<<<B_BATCH, 256, 0, stream>>>(batch_targets, batch_confidences,
                                                  selected_mask, confidences, last_pos);
  sort_kernel<<<C_CLASSES, 256, 0, stream>>>(confidences, batch_confidences, last_pos, order);
  gather_tdm_kernel<<<C_CLASSES, 32, 0, stream>>>(memory, batch_features, last_pos, order, out);
}